// Low_to_high_layer_61366492725289
// MI455X (gfx1250) — compile-verified
//
#include <hip/hip_runtime.h>

typedef float v2f __attribute__((ext_vector_type(2)));
typedef float v8f __attribute__((ext_vector_type(8)));

#define D 128
#define NEG_SLOPE 0.2f

// ---------- order-preserving float <-> uint mapping for atomic max ----------
__device__ __forceinline__ unsigned fmapu(float x) {
    unsigned b = __float_as_uint(x);
    return b ^ ((unsigned)(((int)b) >> 31) | 0x80000000u);
}
__device__ __forceinline__ float funmap(unsigned u) {
    unsigned b = (u & 0x80000000u) ? (u ^ 0x80000000u) : ~u;
    return __uint_as_float(b);
}

__device__ __forceinline__ void atomicAddF(float* p, float v) {
    __hip_atomic_fetch_add(p, v, __ATOMIC_RELAXED, __HIP_MEMORY_SCOPE_AGENT);
}

// ---------------- K0: init scratch + output ----------------
__global__ __launch_bounds__(256) void k_init(float* __restrict__ out,
                                              unsigned* __restrict__ mmax,
                                              float* __restrict__ denom,
                                              float* __restrict__ deg,
                                              int nnodes) {
    int idx = blockIdx.x * 256 + threadIdx.x;
    int total = nnodes * D;
    if (idx < total) out[idx] = 0.0f;
    if (idx < nnodes) {
        mmax[idx]  = 0u;     // below map(-inf): only read for nodes with >=1 edge
        denom[idx] = 0.0f;
        deg[idx]   = 0.0f;
    }
}

// ---------------- K1: Y = X @ W via V_WMMA_F32_16X16X4_F32 ----------------
// One wave computes a 16(row) x 128(col) strip of Y. W (128x128, 64KB) staged in LDS.
__global__ __launch_bounds__(256) void k_gemm(const float* __restrict__ X,
                                              const float* __restrict__ W,
                                              float* __restrict__ Y,
                                              int nrows) {
    __shared__ float lW[D * D];  // 64 KB
    const int t = threadIdx.x;
    // cooperative load of W into LDS (float4)
    for (int i = t; i < D * D / 4; i += 256) {
        ((float4*)lW)[i] = ((const float4*)W)[i];
    }
    __syncthreads();

    const int wave = t >> 5;
    const int lane = t & 31;
    const int tiles = (nrows + 15) >> 4;
    const int mt = blockIdx.x * 8 + wave;   // 16-row tile index
    if (mt >= tiles) return;                // whole-wave uniform exit (EXEC stays full)

    const int m     = lane & 15;            // row within tile (A) / col within 16 (B,D)
    const int khalf = (lane >> 4) << 1;     // 0 for lanes 0-15, 2 for lanes 16-31
    int row = mt * 16 + m;
    if (row >= nrows) row = nrows - 1;      // clamp loads, keep EXEC full for WMMA
    const float* __restrict__ xrow = X + (size_t)row * D;

    v8f acc[8];
#pragma unroll
    for (int n = 0; n < 8; ++n)
#pragma unroll
        for (int i = 0; i < 8; ++i) acc[n][i] = 0.0f;

    for (int k = 0; k < D; k += 4) {
        // A tile 16x4: VGPR0 = K = khalf, VGPR1 = K = khalf+1 (per ISA layout)
        v2f a;
        a[0] = xrow[k + khalf];
        a[1] = xrow[k + khalf + 1];
        const float* wbase = lW + (size_t)(k + khalf) * D + m;
#pragma unroll
        for (int n = 0; n < 8; ++n) {
            // B tile 4x16: VGPR0 = (K=khalf, N=n*16+m), VGPR1 = (K=khalf+1, ...)
            v2f b;
            b[0] = wbase[n * 16];
            b[1] = wbase[D + n * 16];
            acc[n] = __builtin_amdgcn_wmma_f32_16x16x4_f32(
                false, a, false, b, (short)0, acc[n], false, false);
        }
    }

    // D tile 16x16: VGPR i holds (M = i + (lane<16?0:8), N = n*16 + (lane&15))
    const int mhi = (lane >> 4) << 3;
#pragma unroll
    for (int n = 0; n < 8; ++n) {
#pragma unroll
        for (int i = 0; i < 8; ++i) {
            int M = mt * 16 + mhi + i;
            if (M < nrows) Y[(size_t)M * D + n * 16 + m] = acc[n][i];
        }
    }
}

// ---------------- K2: per-edge logits + segment max + degree ----------------
__global__ __launch_bounds__(256) void k_edge_logits(const float* __restrict__ xl,
                                                     const float* __restrict__ xr,
                                                     const float* __restrict__ att,
                                                     const int* __restrict__ esrc,
                                                     const int* __restrict__ edst,
                                                     float* __restrict__ elog,
                                                     unsigned* __restrict__ mmax,
                                                     float* __restrict__ deg,
                                                     int nedges) {
    const int e    = blockIdx.x * 8 + (threadIdx.x >> 5);
    const int lane = threadIdx.x & 31;
    if (e >= nedges) return;
    const int s = esrc[e];
    const int d = edst[e];
    const float4 xj = ((const float4*)(xl + (size_t)s * D))[lane];
    const float4 xi = ((const float4*)(xr + (size_t)d * D))[lane];
    const float4 av = ((const float4*)att)[lane];
    float sum = 0.0f, v;
    v = xi.x + xj.x; sum += av.x * (v > 0.0f ? v : NEG_SLOPE * v);
    v = xi.y + xj.y; sum += av.y * (v > 0.0f ? v : NEG_SLOPE * v);
    v = xi.z + xj.z; sum += av.z * (v > 0.0f ? v : NEG_SLOPE * v);
    v = xi.w + xj.w; sum += av.w * (v > 0.0f ? v : NEG_SLOPE * v);
#pragma unroll
    for (int off = 16; off > 0; off >>= 1) sum += __shfl_xor(sum, off, 32);
    if (lane == 0) {
        elog[e] = sum;
        atomicMax(mmax + d, fmapu(sum));
        atomicAddF(deg + d, 1.0f);
    }
}

// ---------------- K3: exp(e - m[dst]) + segment sum ----------------
__global__ __launch_bounds__(256) void k_edge_exp(const float* __restrict__ elog,
                                                  const unsigned* __restrict__ mmax,
                                                  const int* __restrict__ edst,
                                                  float* __restrict__ ex,
                                                  float* __restrict__ denom,
                                                  int nedges) {
    int e = blockIdx.x * 256 + threadIdx.x;
    if (e >= nedges) return;
    int d = edst[e];
    float m = funmap(mmax[d]);
    float v = expf(elog[e] - m);
    ex[e] = v;
    atomicAddF(denom + d, v);
}

// ---------------- K4: scatter alpha * xj into out ----------------
__global__ __launch_bounds__(256) void k_edge_scatter(const float* __restrict__ xl,
                                                      const int* __restrict__ esrc,
                                                      const int* __restrict__ edst,
                                                      const float* __restrict__ ex,
                                                      const float* __restrict__ denom,
                                                      float* __restrict__ out,
                                                      int nedges) {
    const int e    = blockIdx.x * 8 + (threadIdx.x >> 5);
    const int lane = threadIdx.x & 31;
    if (e >= nedges) return;
    const int s = esrc[e];
    const int d = edst[e];
    const float alpha = ex[e] / fmaxf(denom[d], 1e-16f);
    const float4 xj = ((const float4*)(xl + (size_t)s * D))[lane];
    float* o = out + (size_t)d * D + lane * 4;
    atomicAddF(o + 0, alpha * xj.x);
    atomicAddF(o + 1, alpha * xj.y);
    atomicAddF(o + 2, alpha * xj.z);
    atomicAddF(o + 3, alpha * xj.w);
}

// ---------------- K5: mean + bias ----------------
__global__ __launch_bounds__(256) void k_finalize(float* __restrict__ out,
                                                  const float* __restrict__ deg,
                                                  const float* __restrict__ bias,
                                                  int nnodes) {
    int idx = blockIdx.x * 256 + threadIdx.x;
    if (idx >= nnodes * D) return;
    int n = idx >> 7;
    int dd = idx & (D - 1);
    out[idx] = out[idx] / fmaxf(deg[n], 1.0f) + bias[dd];
}

extern "C" void kernel_launch(void* const* d_in, const int* in_sizes, int n_in,
                              void* d_out, int out_size, void* d_ws, size_t ws_size,
                              hipStream_t stream) {
    const float* x_low  = (const float*)d_in[0];
    const float* x_high = (const float*)d_in[1];
    const float* W_l    = (const float*)d_in[2];
    const float* W_r    = (const float*)d_in[3];
    const float* att    = (const float*)d_in[4];
    const float* bias   = (const float*)d_in[5];
    const int*   esrc   = (const int*)d_in[6];
    const int*   edst   = (const int*)d_in[7];
    const int nnodes = in_sizes[0] / D;
    const int nedges = in_sizes[6];
    float* out = (float*)d_out;

    // workspace carve-up
    char* ws = (char*)d_ws;
    float* xl = (float*)ws;            ws += sizeof(float) * (size_t)nnodes * D;
    float* xr = (float*)ws;            ws += sizeof(float) * (size_t)nnodes * D;
    float* elog = (float*)ws;          ws += sizeof(float) * (size_t)nedges;
    float* ex = (float*)ws;            ws += sizeof(float) * (size_t)nedges;
    unsigned* mmax = (unsigned*)ws;    ws += sizeof(unsigned) * (size_t)nnodes;
    float* denom = (float*)ws;         ws += sizeof(float) * (size_t)nnodes;
    float* deg = (float*)ws;           ws += sizeof(float) * (size_t)nnodes;

    // K0: init
    {
        int total = nnodes * D;
        k_init<<<(total + 255) / 256, 256, 0, stream>>>(out, mmax, denom, deg, nnodes);
    }
    // K1: GEMMs (WMMA f32 16x16x4)
    {
        int tiles = (nnodes + 15) / 16;
        int blocks = (tiles + 7) / 8;
        k_gemm<<<blocks, 256, 0, stream>>>(x_low, W_l, xl, nnodes);
        k_gemm<<<blocks, 256, 0, stream>>>(x_high, W_r, xr, nnodes);
    }
    // K2: logits + segment max + degree
    {
        int blocks = (nedges + 7) / 8;
        k_edge_logits<<<blocks, 256, 0, stream>>>(xl, xr, att, esrc, edst,
                                                  elog, mmax, deg, nedges);
    }
    // K3: exp + denom
    k_edge_exp<<<(nedges + 255) / 256, 256, 0, stream>>>(elog, mmax, edst, ex, denom, nedges);
    // K4: scatter weighted messages
    k_edge_scatter<<<(nedges + 7) / 8, 256, 0, stream>>>(xl, esrc, edst, ex, denom, out, nedges);
    // K5: mean + bias
    k_finalize<<<(nnodes * D + 255) / 256, 256, 0, stream>>>(out, deg, bias, nnodes);
}